// FrFTLayer_40114994545404
// MI455X (gfx1250) — compile-verified
//
#include <hip/hip_runtime.h>
#include <math.h>

typedef __attribute__((ext_vector_type(2))) float v2f;
typedef __attribute__((ext_vector_type(8))) float v8f;

static __device__ __forceinline__ double wrap2pi(double x) {
  const double twopi = 6.28318530717958647692;
  return fmod(x, twopi);
}

#define DPI 3.14159265358979323846

// ---------------------------------------------------------------------------
// Kernel 1: Dirichlet half-band interpolation kernel table (period 4096)
//   D(d) = (-1)^d * cot(pi*(2d+1)/8192) / 4096
// ---------------------------------------------------------------------------
__global__ void k_dirichlet(float* __restrict__ D) {
  int d = blockIdx.x * blockDim.x + threadIdx.x;   // 0..4095
  double th = DPI * (double)(2 * d + 1) / 8192.0;  // in (0, pi) -> no singularity
  double ct = cos(th) / sin(th);
  float sgn = (d & 1) ? -1.0f : 1.0f;
  D[d] = (float)(ct / 4096.0) * sgn;
}

// ---------------------------------------------------------------------------
// Kernel 2: bandlimited 2x interpolation + chirp modulation
//   t even: biz = x[m];  t odd: biz = sum_m' x[m'] * D((m-m') & 4095)
//   z[t] = biz[t] * exp(i*psi_t),
//   psi_t = -pi*tan(phi/2)*((t-4096)/128)^2 + pi*t^2/(16384*sin(phi))
// ---------------------------------------------------------------------------
__global__ void __launch_bounds__(256)
k_interp_mod(const float* __restrict__ x, const float* __restrict__ ord,
             const float* __restrict__ Dt,
             float* __restrict__ Zre, float* __restrict__ Zim) {
  __shared__ float xs[4096];
  __shared__ float ds[4096];
  const int r = blockIdx.y;
  const int tid = threadIdx.x;
  for (int i = tid; i < 4096; i += 256) {
    xs[i] = x[r * 4096 + i];
    ds[i] = Dt[i];
  }
  __syncthreads();
  const int m = blockIdx.x * 256 + tid;  // 0..4095
  float acc = 0.0f;
#pragma unroll 8
  for (int mp = 0; mp < 4096; ++mp)
    acc += xs[mp] * ds[(m - mp) & 4095];

  double a = (double)ord[0];
  double phi = a * DPI * 0.5;
  double s = sin(phi);
  double tp2 = tan(phi * 0.5);
#pragma unroll
  for (int which = 0; which < 2; ++which) {
    int t = 2 * m + which;
    float v = which ? acc : xs[m];
    double u = ((double)t - 4096.0) / 128.0;
    double psi = -DPI * tp2 * u * u + DPI * (double)t * (double)t / (16384.0 * s);
    psi = wrap2pi(psi);
    Zre[(size_t)r * 8192 + t] = v * (float)cos(psi);
    Zim[(size_t)r * 8192 + t] = v * (float)sin(psi);
  }
}

// ---------------------------------------------------------------------------
// Kernel 3: scaled-DFT (chirp-Z core) as WMMA f32 complex matmul
//   out[r,k] = c1[k] * sum_t z[r,t] * exp(-i * c * k * t),  c = pi/(4096*sin(phi))
// Block: 256 thr (8 waves). Wave -> 16x16 complex tile. LDS Z tile 16x64 (pad 68).
// Twiddles: exact f64-reduced phasor per 64-chunk, complex-rotation recurrence.
// ---------------------------------------------------------------------------
__global__ void __launch_bounds__(256)
k_chirpz_wmma(const float* __restrict__ Zre, const float* __restrict__ Zim,
              const float* __restrict__ ord, float* __restrict__ out) {
  __shared__ float zre[16 * 68];
  __shared__ float zim[16 * 68];
  const int tid = threadIdx.x;
  const int wave = tid >> 5, lane = tid & 31;
  const int rowbase = blockIdx.y * 16;
  const int col = (blockIdx.x * 8 + wave) * 16 + (lane & 15);  // output index k
  const int ks = (lane >> 4) * 2;                              // K slot 0/2 per B layout

  double a = (double)ord[0];
  double phi = a * DPI * 0.5;
  double s = sin(phi);
  double c = DPI / (4096.0 * s);

  // per-lane step rotor  e^{-i*4*c*k}
  double d4 = wrap2pi(4.0 * c * (double)col);
  const float dre = (float)cos(d4), dim = -(float)sin(d4);

  v8f acc_re = {};
  v8f acc_im = {};

  for (int tb = 0; tb < 8192; tb += 64) {
    for (int i = tid; i < 1024; i += 256) {
      int rr = i >> 6, tt = i & 63;
      size_t g = (size_t)(rowbase + rr) * 8192 + tb + tt;
      zre[rr * 68 + tt] = Zre[g];
      zim[rr * 68 + tt] = Zim[g];
    }
    if (tb + 64 < 8192) {  // pull next chunk toward L2/L0 (global_prefetch_b8)
      __builtin_prefetch(&Zre[(size_t)(rowbase + (tid >> 4)) * 8192 + tb + 64 + 4 * (tid & 15)], 0, 1);
      __builtin_prefetch(&Zim[(size_t)(rowbase + (tid >> 4)) * 8192 + tb + 64 + 4 * (tid & 15)], 0, 1);
    }
    __syncthreads();

    // exact phasors e^{-i*c*k*t} at t = tb+ks, tb+ks+1  (k*t <= 3.4e7: exact in f64)
    long kt0 = (long)col * (long)(tb + ks);
    double th0 = wrap2pi(c * (double)kt0);
    double th1 = wrap2pi(c * (double)(kt0 + col));
    float g0r = (float)cos(th0), g0i = -(float)sin(th0);
    float g1r = (float)cos(th1), g1i = -(float)sin(th1);

    const int arow = (lane & 15) * 68;
#pragma unroll
    for (int kk = 0; kk < 64; kk += 4) {
      v2f ar = *(const v2f*)&zre[arow + kk + ks];
      v2f ai = *(const v2f*)&zim[arow + kk + ks];
      v2f nai = -ai;
      v2f br; br.x = g0r; br.y = g1r;
      v2f bi; bi.x = g0i; bi.y = g1i;
      // complex MAC: 4 real 16x16x4 f32 WMMAs
      acc_re = __builtin_amdgcn_wmma_f32_16x16x4_f32(false, ar,  false, br, (short)0, acc_re, false, false);
      acc_re = __builtin_amdgcn_wmma_f32_16x16x4_f32(false, nai, false, bi, (short)0, acc_re, false, false);
      acc_im = __builtin_amdgcn_wmma_f32_16x16x4_f32(false, ar,  false, bi, (short)0, acc_im, false, false);
      acc_im = __builtin_amdgcn_wmma_f32_16x16x4_f32(false, ai,  false, br, (short)0, acc_im, false, false);
      // advance phasors by e^{-i*4ck}
      float n0r = g0r * dre - g0i * dim; g0i = g0r * dim + g0i * dre; g0r = n0r;
      float n1r = g1r * dre - g1i * dim; g1i = g1r * dim + g1i * dre; g1r = n1r;
    }
    __syncthreads();
  }

  // epilogue: c1[k] = (Aphi/128) * exp(i*(-pi*tan(phi/2)*u^2 + pi*k^2/(4096 s))), u=(2k-4096)/128
  double tp2 = tan(phi * 0.5);
  double u = (2.0 * (double)col - 4096.0) / 128.0;
  double chi = -DPI * tp2 * u * u + DPI * (double)col * (double)col / (4096.0 * s)
               - (DPI * 0.25 * (s >= 0.0 ? 1.0 : -1.0) - phi * 0.5);
  chi = wrap2pi(chi);
  double amp = 1.0 / (128.0 * sqrt(fabs(s)));
  if (col == 0) amp *= 2.0;  // res[...,0] *= 2
  const float c1r = (float)(amp * cos(chi)), c1i = (float)(amp * sin(chi));

#pragma unroll
  for (int g = 0; g < 8; ++g) {
    int row = rowbase + g + 8 * (lane >> 4);  // C/D layout: M = g + 8*(lane>=16)
    float re = acc_re[g], im = acc_im[g];
    size_t idx = ((size_t)row * 4096 + col) * 2;  // complex64 interleaved
    out[idx]     = c1r * re - c1i * im;
    out[idx + 1] = c1r * im + c1i * re;
  }
}

// ---------------------------------------------------------------------------
extern "C" void kernel_launch(void* const* d_in, const int* in_sizes, int n_in,
                              void* d_out, int out_size, void* d_ws, size_t ws_size,
                              hipStream_t stream) {
  const float* x   = (const float*)d_in[0];
  const float* ord = (const float*)d_in[1];
  float* ws  = (float*)d_ws;
  float* D   = ws;                         // 4096 f32
  float* Zre = ws + 4096;                  // 128*8192 f32
  float* Zim = Zre + 128 * 8192;           // 128*8192 f32   (total ~8.4 MB)

  k_dirichlet<<<16, 256, 0, stream>>>(D);
  k_interp_mod<<<dim3(16, 128), 256, 0, stream>>>(x, ord, D, Zre, Zim);
  k_chirpz_wmma<<<dim3(32, 8), 256, 0, stream>>>(Zre, Zim, ord, (float*)d_out);
}